// EdgeDegreeEmbedding_82214263980367
// MI455X (gfx1250) — compile-verified
//
#include <hip/hip_runtime.h>
#include <hip/hip_bf16.h>
#include <math.h>

typedef __attribute__((ext_vector_type(2))) float v2f;
typedef __attribute__((ext_vector_type(8))) float v8f;

#define E_TOT    80000
#define N_NODES  4000
#define HID      128
#define MAX_ELEM 90
#define LN_EPS   1e-5f

// shared-memory layout (in floats). x_edge aliases h2 (x dead before h2 written).
#define H2_OFF   0         // 16 x 644
#define H2_STR   644
#define X_OFF    0         // 16 x 388 (aliases H2 region)
#define X_STR    388
#define H1_OFF   10304     // 16 x 132
#define H1_STR   132
#define WIG_OFF  12416     // 16 x 5 x 26
#define SMEM_F   14496     // 57984 bytes

__global__ void edeg_zero_kernel(float* __restrict__ p, int n) {
    int i = blockIdx.x * 256 + threadIdx.x;
    if (i < n) p[i] = 0.0f;
}

// W1 (384x128) -> W1T (128 rows x 384); W2 (128x640) -> W2T (640 rows x 128)
__global__ void edeg_transpose_kernel(const float* __restrict__ W1,
                                      const float* __restrict__ W2,
                                      float* __restrict__ w1t,
                                      float* __restrict__ w2t) {
    int i = blockIdx.x * 256 + threadIdx.x;
    if (i < 384 * 128) {
        int k = i / 128, n = i % 128;
        w1t[n * 384 + k] = W1[i];
    }
    if (i < 128 * 640) {
        int k = i / 640, n = i % 640;
        w2t[n * 128 + k] = W2[i];
    }
}

// CDNA5 async global->LDS copies (bypass VGPRs, tracked on ASYNCcnt)
static __device__ __forceinline__ void async_lds_b128(const float* smem_dst,
                                                      const float* gsrc) {
    unsigned l = (unsigned)(uintptr_t)smem_dst;   // low 32 bits = DS byte offset
    asm volatile("global_load_async_to_lds_b128 %0, %1, off"
                 : : "v"(l), "v"(gsrc) : "memory");
}
static __device__ __forceinline__ void async_lds_b32(const float* smem_dst,
                                                     const float* gsrc) {
    unsigned l = (unsigned)(uintptr_t)smem_dst;
    asm volatile("global_load_async_to_lds_b32 %0, %1, off"
                 : : "v"(l), "v"(gsrc) : "memory");
}

__global__ __launch_bounds__(128) void edeg_main_kernel(
    const float* __restrict__ r,
    const float* __restrict__ edt,          // edge_distance_embedding (E,128)
    const int*   __restrict__ edge_index,   // (2,E)
    const float* __restrict__ wigner,       // (E,25,25)
    const float* __restrict__ src_emb,      // (90,128)
    const float* __restrict__ tgt_emb,      // (90,128)
    const float* __restrict__ b1,
    const float* __restrict__ ln_g,
    const float* __restrict__ ln_b,
    const float* __restrict__ b2,
    const float* __restrict__ w1t,          // (128,384)
    const float* __restrict__ w2t,          // (640,128)
    float* __restrict__ out)                // (N,25,128)
{
    __shared__ float smem[SMEM_F];
    __shared__ int   s_src[16], s_tgtc[16], s_tgt[16];
    __shared__ float s_env[16];

    const int t    = threadIdx.x;
    const int e0   = blockIdx.x * 16;
    const int lane = t & 31;
    const int w    = t >> 5;          // wave id 0..3
    const int half = lane >> 4;       // 0/1
    const int lm   = lane & 15;

    // ---- stage 0: per-edge metadata ----
    if (t < 16) {
        int e = e0 + t; if (e >= E_TOT) e = E_TOT - 1;
        int i0 = edge_index[e];
        int i1 = edge_index[E_TOT + e];
        int c0 = i0 < 0 ? 0 : (i0 >= MAX_ELEM ? MAX_ELEM - 1 : i0);  // JAX OOB clamp
        int c1 = i1 < 0 ? 0 : (i1 >= MAX_ELEM ? MAX_ELEM - 1 : i1);
        s_src[t]  = c0;
        s_tgtc[t] = c1;
        s_tgt[t]  = i1;
        float d  = r[e] * (1.0f / 12.0f);
        float d2 = d * d, d4 = d2 * d2, d5 = d4 * d;
        // p=5: 1 - 21 d^5 + 35 d^6 - 15 d^7
        float env = 1.0f + d5 * (-21.0f + d * (35.0f - 15.0f * d));
        s_env[t] = (d < 1.0f ? env : 0.0f) * 0.2f;   // fold in 1/RESCALE
    }
    __syncthreads();

    // ---- stage 1: async-stage x_edge (16x384) and 5 wigner rows/edge into LDS ----
    // x_edge: 16 edges x 3 segments x 128 floats = 1536 b128 chunks
    for (int q = t; q < 1536; q += 128) {
        int i = q / 96, rem = q % 96;
        int seg = rem / 32, c4 = (rem % 32) * 4;
        int e = e0 + i; if (e >= E_TOT) e = E_TOT - 1;
        const float* g;
        if (seg == 0)      g = edt     + (size_t)e * 128        + c4;
        else if (seg == 1) g = src_emb + (size_t)s_src[i] * 128 + c4;
        else               g = tgt_emb + (size_t)s_tgtc[i] * 128 + c4;
        async_lds_b128(&smem[X_OFF + i * X_STR + seg * 128 + c4], g);
    }
    // wigner rows 0,2,6,12,20 of each edge (25 floats each)
    for (int idx = t; idx < 2000; idx += 128) {
        int i = idx / 125, rem = idx % 125;
        int j = rem / 25, f = rem % 25;
        int e = e0 + i; if (e >= E_TOT) e = E_TOT - 1;
        int g = j * (j + 1);                      // wigner row: 0,2,6,12,20
        async_lds_b32(&smem[WIG_OFF + (i * 5 + j) * 26 + f],
                      wigner + (size_t)e * 625 + g * 25 + f);
    }
    asm volatile("s_wait_asynccnt 0" ::: "memory");
    __syncthreads();

    // ---- stage 2: GEMM1  h1 = x_edge(16x384) @ W1(384x128) + b1  (f32 WMMA) ----
    {
        v8f acc0 = {};
        v8f acc1 = {};
        const float* brow0 = w1t + (size_t)(16 * w + lm) * 384 + 2 * half;
        const float* brow1 = w1t + (size_t)(16 * (w + 4) + lm) * 384 + 2 * half;
        const float* arow  = &smem[X_OFF + lm * X_STR + 2 * half];
        #pragma unroll 4
        for (int k = 0; k < 96; ++k) {
            v2f a  = *(const v2f*)(arow  + 4 * k);
            v2f bA = *(const v2f*)(brow0 + 4 * k);
            v2f bB = *(const v2f*)(brow1 + 4 * k);
            acc0 = __builtin_amdgcn_wmma_f32_16x16x4_f32(false, a, false, bA,
                                                         (short)0, acc0, false, false);
            acc1 = __builtin_amdgcn_wmma_f32_16x16x4_f32(false, a, false, bB,
                                                         (short)0, acc1, false, false);
        }
        float bias0 = b1[16 * w + lm];
        float bias1 = b1[16 * (w + 4) + lm];
        #pragma unroll
        for (int v = 0; v < 8; ++v) {
            int row = v + 8 * half;
            smem[H1_OFF + row * H1_STR + 16 * w + lm]       = acc0[v] + bias0;
            smem[H1_OFF + row * H1_STR + 16 * (w + 4) + lm] = acc1[v] + bias1;
        }
    }
    __syncthreads();

    // ---- stage 3: LayerNorm + SiLU (8 threads per edge-row) ----
    {
        int row = t >> 3, s = t & 7;
        float* hp = &smem[H1_OFF + row * H1_STR + s * 16];
        float sum = 0.0f, sq = 0.0f;
        #pragma unroll
        for (int q = 0; q < 16; ++q) { float x = hp[q]; sum += x; sq += x * x; }
        #pragma unroll
        for (int m = 1; m < 8; m <<= 1) {
            sum += __shfl_xor(sum, m, 8);
            sq  += __shfl_xor(sq,  m, 8);
        }
        float mu  = sum * (1.0f / 128.0f);
        float var = sq * (1.0f / 128.0f) - mu * mu;
        float rs  = rsqrtf(var + LN_EPS);
        #pragma unroll
        for (int q = 0; q < 16; ++q) {
            int c = s * 16 + q;
            float x = (hp[q] - mu) * rs * ln_g[c] + ln_b[c];
            hp[q] = x / (1.0f + __expf(-x));        // SiLU
        }
    }
    __syncthreads();

    // ---- stage 4: GEMM2  h2 = act(16x128) @ W2(128x640) + b2  (f32 WMMA) ----
    for (int g = 0; g < 10; ++g) {
        int ct = 4 * g + w;                         // column tile 0..39
        v8f acc = {};
        const float* brow = w2t + (size_t)(16 * ct + lm) * 128 + 2 * half;
        const float* arow = &smem[H1_OFF + lm * H1_STR + 2 * half];
        #pragma unroll 8
        for (int k = 0; k < 32; ++k) {
            v2f a  = *(const v2f*)(arow + 4 * k);
            v2f bb = *(const v2f*)(brow + 4 * k);
            acc = __builtin_amdgcn_wmma_f32_16x16x4_f32(false, a, false, bb,
                                                        (short)0, acc, false, false);
        }
        float bias = b2[16 * ct + lm];
        #pragma unroll
        for (int v = 0; v < 8; ++v)
            smem[H2_OFF + (v + 8 * half) * H2_STR + 16 * ct + lm] = acc[v] + bias;
    }
    __syncthreads();

    // ---- stage 5: wigner contraction + envelope + scatter-add (c = t) ----
    for (int i = 0; i < 16; ++i) {
        int e = e0 + i; if (e >= E_TOT) continue;
        float h0 = smem[H2_OFF + i * H2_STR + 0 * 128 + t];
        float h1 = smem[H2_OFF + i * H2_STR + 1 * 128 + t];
        float h2 = smem[H2_OFF + i * H2_STR + 2 * 128 + t];
        float h3 = smem[H2_OFF + i * H2_STR + 3 * 128 + t];
        float h4 = smem[H2_OFF + i * H2_STR + 4 * 128 + t];
        float es = s_env[i];
        const float* wp = &smem[WIG_OFF + i * 5 * 26];
        float* op = out + (size_t)s_tgt[i] * 3200 + t;
        #pragma unroll
        for (int f = 0; f < 25; ++f) {
            float acc = wp[f]       * h0 + wp[26 + f]  * h1 +
                        wp[52 + f]  * h2 + wp[78 + f]  * h3 +
                        wp[104 + f] * h4;
            atomicAdd(op + f * 128, acc * es);
        }
    }
}

extern "C" void kernel_launch(void* const* d_in, const int* in_sizes, int n_in,
                              void* d_out, int out_size, void* d_ws, size_t ws_size,
                              hipStream_t stream) {
    (void)in_sizes; (void)n_in; (void)ws_size;
    const float* r        = (const float*)d_in[0];
    // d_in[1] atomic_numbers: unused by reference
    const float* edt      = (const float*)d_in[2];
    const int*   eidx     = (const int*)d_in[3];
    const float* wigner   = (const float*)d_in[4];
    const float* src_emb  = (const float*)d_in[5];
    const float* tgt_emb  = (const float*)d_in[6];
    const float* W1       = (const float*)d_in[7];
    const float* b1       = (const float*)d_in[8];
    const float* ln_g     = (const float*)d_in[9];
    const float* ln_b     = (const float*)d_in[10];
    const float* W2       = (const float*)d_in[11];
    const float* b2       = (const float*)d_in[12];
    // d_in[13] to_m, d_in[14] out_idx: folded into constants (permutation is fixed)
    float* out = (float*)d_out;
    float* w1t = (float*)d_ws;            // 128*384 floats
    float* w2t = w1t + 384 * 128;         // 640*128 floats (total 512 KB scratch)

    edeg_zero_kernel<<<(out_size + 255) / 256, 256, 0, stream>>>(out, out_size);
    edeg_transpose_kernel<<<(128 * 640 + 255) / 256, 256, 0, stream>>>(W1, W2, w1t, w2t);
    edeg_main_kernel<<<E_TOT / 16, 128, 0, stream>>>(
        r, edt, eidx, wigner, src_emb, tgt_emb,
        b1, ln_g, ln_b, b2, w1t, w2t, out);
}